// Attention_83373905150115
// MI455X (gfx1250) — compile-verified
//
#include <hip/hip_runtime.h>
#include <math.h>

typedef __attribute__((ext_vector_type(2))) float v2f;
typedef __attribute__((ext_vector_type(8))) float v8f;

#define BB 32
#define TT 2048
#define RNN_DIM 1024
#define EMB_DIM 512
#define ATT_DIM 128
#define N_FILT 32
#define KSIZE 31
#define PADW 15

// d_out layout (floats): context[32*512] | attn_w[32*2048] | pq[32*128] | attn_aux[32*2048]
#define OUT_CTX   0
#define OUT_ATTN  16384
#define OUT_PQ    81920
#define OUT_ATTNA 86016

// workspace layout (floats)
#define WS_E    0
#define WS_EA   65536
#define WS_PQ   131072
#define WS_PQA  135168

// Hardware tanh (V_TANH_F32 is a CDNA5 TRANS op); branch-free fallback otherwise.
#if defined(__has_builtin)
# if __has_builtin(__builtin_amdgcn_tanhf)
#  define USE_HW_TANH 1
# endif
#endif

__device__ __forceinline__ float fast_tanh(float x) {
#ifdef USE_HW_TANH
    return __builtin_amdgcn_tanhf(x);
#else
    float e = __expf(-2.0f * __builtin_fabsf(x));   // e in (0,1], no overflow
    float t = (1.0f - e) / (1.0f + e);
    return __builtin_copysignf(t, x);
#endif
}

// ---------------------------------------------------------------------------
// Kernel 1: pq = ahs @ wq^T and pq_aux = ahs @ wq_aux^T. One wave per dot.
// ---------------------------------------------------------------------------
__global__ void pq_kernel(const float* __restrict__ ahs,
                          const float* __restrict__ wq,
                          const float* __restrict__ wqa,
                          float* __restrict__ pq_ws,
                          float* __restrict__ pqa_ws,
                          float* __restrict__ pq_out) {
    int gtid = blockIdx.x * blockDim.x + threadIdx.x;
    int wave = gtid >> 5;
    int lane = threadIdx.x & 31;
    int which = wave >> 12;          // 4096 waves per matrix
    int rem   = wave & 4095;
    int b = rem >> 7;
    int a = rem & 127;
    const float* W = which ? wqa : wq;
    const float* arow = ahs + b * RNN_DIM;
    const float* wrow = W + a * RNN_DIM;
    float acc = 0.f;
#pragma unroll 4
    for (int k = lane; k < RNN_DIM; k += 32)
        acc += arow[k] * wrow[k];
#pragma unroll
    for (int m = 16; m >= 1; m >>= 1)
        acc += __shfl_xor(acc, m, 32);
    if (lane == 0) {
        if (which == 0) {
            pq_ws[b * ATT_DIM + a]  = acc;
            pq_out[b * ATT_DIM + a] = acc;   // pq is also a tuple output
        } else {
            pqa_ws[b * ATT_DIM + a] = acc;
        }
    }
}

// ---------------------------------------------------------------------------
// Kernel 2: fused conv1d -> (loc @ w_loc^T) via V_WMMA_F32_16X16X4_F32
//           -> energies = sum_a tanh(pq+ploc+pm)*v, and aux energies.
// Grid: (B, 4) blocks of 256 threads. Each wave handles a 16(t) x 128(a) tile.
// ---------------------------------------------------------------------------
__global__ void energies_kernel(const float* __restrict__ aw,      // (B,2,T)
                                const float* __restrict__ conv_w,  // (32,2,31)
                                const float* __restrict__ w_loc,   // (128,32)
                                const float* __restrict__ pm,      // (B,T,128)
                                const float* __restrict__ pa,      // (B,T,128)
                                const float* __restrict__ pq_ws,   // (B,128)
                                const float* __restrict__ pqa_ws,  // (B,128)
                                const float* __restrict__ v,       // (128)
                                const float* __restrict__ v_aux,   // (128)
                                float* __restrict__ e_ws,          // (B,T)
                                float* __restrict__ ea_ws) {       // (B,T)
    __shared__ float aw_sh[2 * (TT + 2 * PADW)];          // padded conv input
    __shared__ float cw_sh[N_FILT * 2 * KSIZE];
    __shared__ float wl_sh[ATT_DIM * N_FILT];
    __shared__ float pq_sh[ATT_DIM], pqa_sh[ATT_DIM];
    __shared__ float v_sh[ATT_DIM],  va_sh[ATT_DIM];

    const int b   = blockIdx.x;
    const int tid = threadIdx.x;
    const int ROWLEN = TT + 2 * PADW;

    // --- stage LDS: padded conv input row, conv weights, w_loc, pq/v vectors ---
    for (int idx = tid; idx < 2 * ROWLEN; idx += 256) {
        int i  = idx / ROWLEN;
        int tm = (idx % ROWLEN) - PADW;
        aw_sh[idx] = (tm >= 0 && tm < TT) ? aw[(b * 2 + i) * TT + tm] : 0.f;
    }
    for (int idx = tid; idx < N_FILT * 2 * KSIZE; idx += 256) cw_sh[idx] = conv_w[idx];
    for (int idx = tid; idx < ATT_DIM * N_FILT; idx += 256)   wl_sh[idx] = w_loc[idx];
    if (tid < ATT_DIM) {
        pq_sh[tid]  = pq_ws[b * ATT_DIM + tid];
        pqa_sh[tid] = pqa_ws[b * ATT_DIM + tid];
        v_sh[tid]   = v[tid];
        va_sh[tid]  = v_aux[tid];
    }
    __syncthreads();

    const int wave = tid >> 5;
    const int lane = tid & 31;
    const int l15  = lane & 15;
    const int lh   = lane >> 4;      // 0 or 1
    const int fh   = 2 * lh;         // K offset within 4-wide chunk

    for (int it = 0; it < 4; ++it) {
        const int t_base = blockIdx.y * 512 + it * 128 + wave * 16;
        const int t = t_base + l15;  // A-fragment row (M) for this lane

        // --- conv: loc[t][f] for f = 4*kb + fh + {0,1}, kb = 0..7 ---
        float la0[8], la1[8];
#pragma unroll
        for (int kb = 0; kb < 8; ++kb) { la0[kb] = 0.f; la1[kb] = 0.f; }
        for (int k = 0; k < KSIZE; ++k) {
            float x0 = aw_sh[t + k];
            float x1 = aw_sh[ROWLEN + t + k];
#pragma unroll
            for (int kb = 0; kb < 8; ++kb) {
                int f0 = 4 * kb + fh;
                la0[kb] += x0 * cw_sh[(f0 * 2 + 0) * KSIZE + k]
                         + x1 * cw_sh[(f0 * 2 + 1) * KSIZE + k];
                la1[kb] += x0 * cw_sh[((f0 + 1) * 2 + 0) * KSIZE + k]
                         + x1 * cw_sh[((f0 + 1) * 2 + 1) * KSIZE + k];
            }
        }

        float acc[8], acca[8];
#pragma unroll
        for (int r = 0; r < 8; ++r) { acc[r] = 0.f; acca[r] = 0.f; }

        for (int nt = 0; nt < 8; ++nt) {
            const int a = nt * 16 + l15;             // C-layout column for this lane
            v8f c = {0.f, 0.f, 0.f, 0.f, 0.f, 0.f, 0.f, 0.f};
#pragma unroll
            for (int kb = 0; kb < 8; ++kb) {
                v2f av, bv;
                av.x = la0[kb];
                av.y = la1[kb];
                bv.x = wl_sh[a * N_FILT + 4 * kb + fh];       // B[k][n] = w_loc[a][f]
                bv.y = wl_sh[a * N_FILT + 4 * kb + fh + 1];
                c = __builtin_amdgcn_wmma_f32_16x16x4_f32(
                        false, av, false, bv, (short)0, c, false, false);
            }
            const float pqv  = pq_sh[a];
            const float vv   = v_sh[a];
            const float pqav = pqa_sh[a];
            const float vav  = va_sh[a];
            const size_t base = ((size_t)b * TT + t_base + 8 * lh) * ATT_DIM + a;
#pragma unroll
            for (int r = 0; r < 8; ++r) {
                float pmv = pm[base + (size_t)r * ATT_DIM];
                float pav = pa[base + (size_t)r * ATT_DIM];
                acc[r]  += fast_tanh(c[r] + pqv + pmv) * vv;
                acca[r] += fast_tanh(pqav + pav) * vav;
            }
        }

        // reduce over a (16 lanes of each half-wave hold distinct columns)
#pragma unroll
        for (int r = 0; r < 8; ++r) {
#pragma unroll
            for (int m = 1; m < 16; m <<= 1) {
                acc[r]  += __shfl_xor(acc[r],  m, 32);
                acca[r] += __shfl_xor(acca[r], m, 32);
            }
        }
        if (l15 == 0) {
#pragma unroll
            for (int r = 0; r < 8; ++r) {
                e_ws [b * TT + t_base + 8 * lh + r] = acc[r];
                ea_ws[b * TT + t_base + 8 * lh + r] = acca[r];
            }
        }
    }
}

// ---------------------------------------------------------------------------
// Kernel 3: masked softmax over T. Grid (B, 2): y=0 -> attn_w, y=1 -> attn_aux.
// ---------------------------------------------------------------------------
__global__ void softmax_kernel(const float* __restrict__ e_ws,
                               const float* __restrict__ ea_ws,
                               const unsigned char* __restrict__ mask,
                               const unsigned char* __restrict__ mask_aux,
                               float* __restrict__ out) {
    const int b  = blockIdx.x;
    const int br = blockIdx.y;
    const float* e = (br ? ea_ws : e_ws) + b * TT;
    const unsigned char* m = (br ? mask_aux : mask) + b * TT;
    float* o = out + (br ? OUT_ATTNA : OUT_ATTN) + b * TT;

    __shared__ float sred[8];
    const int tid = threadIdx.x;           // 256 threads, 8 elems each
    float vals[8];
    float mx = -__builtin_inff();
#pragma unroll
    for (int i = 0; i < 8; ++i) {
        int tt = tid + i * 256;
        float x = m[tt] ? -__builtin_inff() : e[tt];
        vals[i] = x;
        mx = fmaxf(mx, x);
    }
#pragma unroll
    for (int s = 16; s >= 1; s >>= 1) mx = fmaxf(mx, __shfl_xor(mx, s, 32));
    if ((tid & 31) == 0) sred[tid >> 5] = mx;
    __syncthreads();
    if (tid == 0) {
        float x = sred[0];
#pragma unroll
        for (int i = 1; i < 8; ++i) x = fmaxf(x, sred[i]);
        sred[0] = x;
    }
    __syncthreads();
    mx = sred[0];
    __syncthreads();

    float sum = 0.f;
#pragma unroll
    for (int i = 0; i < 8; ++i) {
        vals[i] = __expf(vals[i] - mx);
        sum += vals[i];
    }
#pragma unroll
    for (int s = 16; s >= 1; s >>= 1) sum += __shfl_xor(sum, s, 32);
    if ((tid & 31) == 0) sred[tid >> 5] = sum;
    __syncthreads();
    if (tid == 0) {
        float x = 0.f;
#pragma unroll
        for (int i = 0; i < 8; ++i) x += sred[i];
        sred[0] = x;
    }
    __syncthreads();
    const float inv = 1.0f / sred[0];
#pragma unroll
    for (int i = 0; i < 8; ++i) o[tid + i * 256] = vals[i] * inv;
}

// ---------------------------------------------------------------------------
// Kernel 4: context[b,d] = sum_t attn_w*memory + sum_t attn_aux*memory_aux.
// Grid (B, 8): each block streams a 256-long t chunk; float2 (b64) loads,
// prefetch 8 rows ahead, t unrolled x2 for deeper HBM pipelining.
// ---------------------------------------------------------------------------
__global__ void context_kernel(const float* __restrict__ mem,
                               const float* __restrict__ mema,
                               const float* __restrict__ out_attn,   // = d_out
                               float* __restrict__ ctx) {            // = d_out ctx region
    const int b     = blockIdx.x;
    const int chunk = blockIdx.y;
    const int tid   = threadIdx.x;   // 256 threads, one float2 (2 d's) each
    const float* awr = out_attn + OUT_ATTN  + b * TT + chunk * 256;
    const float* aar = out_attn + OUT_ATTNA + b * TT + chunk * 256;
    const float* mr  = mem  + ((size_t)b * TT + chunk * 256) * EMB_DIM;
    const float* mar = mema + ((size_t)b * TT + chunk * 256) * EMB_DIM;
    float ax = 0.f, ay = 0.f;
#pragma unroll 2
    for (int tt = 0; tt < 256; ++tt) {
        const float2* m0 = (const float2*)(mr  + (size_t)tt * EMB_DIM);
        const float2* m1 = (const float2*)(mar + (size_t)tt * EMB_DIM);
        // keep the HBM stream deep: prefetch 8 rows (16 KB) ahead
        __builtin_prefetch((const char*)m0 + 8 * EMB_DIM * sizeof(float), 0, 1);
        __builtin_prefetch((const char*)m1 + 8 * EMB_DIM * sizeof(float), 0, 1);
        float w  = awr[tt];
        float wa = aar[tt];
        float2 x0 = m0[tid];
        float2 x1 = m1[tid];
        ax += w * x0.x + wa * x1.x;
        ay += w * x0.y + wa * x1.y;
    }
    atomicAdd(&ctx[b * EMB_DIM + 2 * tid],     ax);
    atomicAdd(&ctx[b * EMB_DIM + 2 * tid + 1], ay);
}

// ---------------------------------------------------------------------------
extern "C" void kernel_launch(void* const* d_in, const int* in_sizes, int n_in,
                              void* d_out, int out_size, void* d_ws, size_t ws_size,
                              hipStream_t stream) {
    const float* ahs   = (const float*)d_in[0];
    const float* mem   = (const float*)d_in[1];
    const float* pm    = (const float*)d_in[2];
    const float* aw    = (const float*)d_in[3];
    const float* mema  = (const float*)d_in[4];
    const float* pa    = (const float*)d_in[5];
    const unsigned char* mask  = (const unsigned char*)d_in[6];
    const unsigned char* maska = (const unsigned char*)d_in[7];
    const float* wq    = (const float*)d_in[8];
    const float* wqa   = (const float*)d_in[9];
    const float* v     = (const float*)d_in[10];
    const float* vaux  = (const float*)d_in[11];
    const float* convw = (const float*)d_in[12];
    const float* wloc  = (const float*)d_in[13];

    float* out = (float*)d_out;
    float* ws  = (float*)d_ws;
    float* e_ws   = ws + WS_E;
    float* ea_ws  = ws + WS_EA;
    float* pq_ws  = ws + WS_PQ;
    float* pqa_ws = ws + WS_PQA;

    // zero the context accumulation region (atomicAdd target)
    hipMemsetAsync(out + OUT_CTX, 0, (size_t)BB * EMB_DIM * sizeof(float), stream);

    pq_kernel<<<1024, 256, 0, stream>>>(ahs, wq, wqa, pq_ws, pqa_ws, out + OUT_PQ);

    energies_kernel<<<dim3(BB, 4), 256, 0, stream>>>(
        aw, convw, wloc, pm, pa, pq_ws, pqa_ws, v, vaux, e_ws, ea_ws);

    softmax_kernel<<<dim3(BB, 2), 256, 0, stream>>>(e_ws, ea_ws, mask, maska, out);

    context_kernel<<<dim3(BB, 8), 256, 0, stream>>>(mem, mema, out, out + OUT_CTX);
}